// MeshLaplacianLoss_65472481460554
// MI455X (gfx1250) — compile-verified
//
#include <hip/hip_runtime.h>

typedef __attribute__((ext_vector_type(2))) float v2f;
typedef __attribute__((ext_vector_type(8))) float v8f;

#define NV     250000
#define NB     16
#define NF     500000
#define NELEM  (NB * NV * 3)          /* 12,000,000 */
#define NSCAT  (NF * 6 * NB)          /* 48,000,000 */

/* workspace layout, in floats:
   [0, 12M)            acc   (scatter accumulator, [v][b*3])
   [12M, 24M)          diff  (v1 - v2, transposed to [v][b*3])
   [24M, 24.25M)       deg -> rdeg
   [24.25M, +2)        double sum accumulator (byte off 97,000,000, 8B aligned) */
#define ACC_OFF   0
#define DIFF_OFF  12000000
#define DEG_OFF   24000000
#define SUM_OFF   24250000
#define ZERO_N    12250002            /* acc + deg + sum slots */

__global__ void mll_zero(float* __restrict__ ws) {
    int t = blockIdx.x * blockDim.x + threadIdx.x;
    if (t < ZERO_N) {
        int idx = (t < NELEM) ? t : (t + 12000000);  /* skip diff region */
        ws[idx] = 0.0f;
    }
}

/* diff[v*48 + b*3 + c] = v1[b][v][c] - v2[b][v][c]; reads fully coalesced */
__global__ void mll_diff(const float* __restrict__ v1, const float* __restrict__ v2,
                         float* __restrict__ ws) {
    int t = blockIdx.x * blockDim.x + threadIdx.x;
    if (t < NELEM) {
        int b = t / (NV * 3);
        int r = t - b * (NV * 3);
        int v = r / 3;
        int c = r - 3 * v;
        ws[DIFF_OFF + v * 48 + b * 3 + c] = v1[t] - v2[t];
    }
}

/* each face contributes exactly 2 directed edges out of each of its vertices */
__global__ void mll_deg(const int* __restrict__ faces, float* __restrict__ ws) {
    int f = blockIdx.x * blockDim.x + threadIdx.x;
    if (f < NF) {
        float* deg = ws + DEG_OFF;
        atomicAdd(deg + faces[3 * f + 0], 2.0f);
        atomicAdd(deg + faces[3 * f + 1], 2.0f);
        atomicAdd(deg + faces[3 * f + 2], 2.0f);
    }
}

__global__ void mll_rdeg(float* __restrict__ ws) {
    int v = blockIdx.x * blockDim.x + threadIdx.x;
    if (v < NV) {
        float* deg = ws + DEG_OFF;
        deg[v] = 1.0f / fmaxf(deg[v], 1.0f);
    }
}

/* one thread per (directed edge, batch); 16 consecutive threads share one edge
   -> contiguous 192B gather from diff[c] and contiguous atomics into acc[r] */
__global__ void mll_scatter(const int* __restrict__ faces, float* __restrict__ ws) {
    int t = blockIdx.x * blockDim.x + threadIdx.x;
    if (t >= NSCAT) return;
    int b = t & 15;
    int e = t >> 4;              /* directed edge id in [0, 3M) */
    int f = e / 6;
    int p = e - 6 * f;           /* pair index 0..5 */

    int vi = faces[3 * f + 0];
    int vj = faces[3 * f + 1];
    int vk = faces[3 * f + 2];

    /* even p: (q, q+1 mod 3); odd p: swapped.  q = p/2 */
    int q  = p >> 1;
    int r0 = q;
    int c0 = (q == 2) ? 0 : (q + 1);
    if (p & 1) { int tmp = r0; r0 = c0; c0 = tmp; }
    int rv = (r0 == 0) ? vi : ((r0 == 1) ? vj : vk);
    int cv = (c0 == 0) ? vi : ((c0 == 1) ? vj : vk);

    const float* src = ws + DIFF_OFF + cv * 48 + b * 3;
    float*       dst = ws + ACC_OFF  + rv * 48 + b * 3;
    atomicAdd(dst + 0, src[0]);
    atomicAdd(dst + 1, src[1]);
    atomicAdd(dst + 2, src[2]);
}

/* grid-stride abs-diff reduction; wave-level sum via V_WMMA_F32_16X16X4_F32
   with B = ones: D[m][n] = s[m] + s[m+16] exactly in fp32.               */
__global__ void mll_reduce(float* __restrict__ ws) {
    const float* acc  = ws + ACC_OFF;
    const float* diff = ws + DIFF_OFF;
    const float* rdeg = ws + DEG_OFF;

    float s = 0.0f;
    int stride = gridDim.x * blockDim.x;
    for (int i = blockIdx.x * blockDim.x + threadIdx.x; i < NELEM; i += stride) {
        int v = i / 48;
        s += fabsf(fmaf(acc[i], rdeg[v], -diff[i]));
    }

    /* A: 16x4 f32, lane L vgpr0 = K (0 or 2), vgpr1 = K+1 (zeroed).
       B = all ones (layout independent).  EXEC is all ones here.        */
    v2f A; A.x = s;    A.y = 0.0f;
    v2f B; B.x = 1.0f; B.y = 1.0f;
    v8f C = {};
    C = __builtin_amdgcn_wmma_f32_16x16x4_f32(
            /*neg_a=*/false, A, /*neg_b=*/false, B,
            /*c_mod=*/(short)0, C, /*reuse_a=*/false, /*reuse_b=*/false);

    float t = C[0] + C[1] + C[2] + C[3] + C[4] + C[5] + C[6] + C[7];
    unsigned lane = threadIdx.x & 31u;
    if (lane == 0u || lane == 16u) {
        double* sum = (double*)(ws + SUM_OFF);
        atomicAdd(sum, (double)t);   /* global_atomic_add_f64 */
    }
}

__global__ void mll_finalize(const float* __restrict__ ws, float* __restrict__ out) {
    if (threadIdx.x == 0 && blockIdx.x == 0) {
        const double* sum = (const double*)(ws + SUM_OFF);
        out[0] = (float)(sum[0] / (double)NELEM);
    }
}

extern "C" void kernel_launch(void* const* d_in, const int* in_sizes, int n_in,
                              void* d_out, int out_size, void* d_ws, size_t ws_size,
                              hipStream_t stream) {
    const float* v1    = (const float*)d_in[0];
    const float* v2    = (const float*)d_in[1];
    const int*   faces = (const int*)d_in[2];
    float*       out   = (float*)d_out;
    float*       ws    = (float*)d_ws;

    const int BS = 256;
    mll_zero   <<<(ZERO_N + BS - 1) / BS, BS, 0, stream>>>(ws);
    mll_diff   <<<(NELEM  + BS - 1) / BS, BS, 0, stream>>>(v1, v2, ws);
    mll_deg    <<<(NF     + BS - 1) / BS, BS, 0, stream>>>(faces, ws);
    mll_rdeg   <<<(NV     + BS - 1) / BS, BS, 0, stream>>>(ws);
    mll_scatter<<<(NSCAT  + BS - 1) / BS, BS, 0, stream>>>(faces, ws);
    mll_reduce <<<4096, BS, 0, stream>>>(ws);
    mll_finalize<<<1, 32, 0, stream>>>(ws, out);
}